// CompLayer_37512244363763
// MI455X (gfx1250) — compile-verified
//
#include <hip/hip_runtime.h>
#include <math.h>

// CDNA5 / gfx1250: wave32. D = 128 for this layer.
#define DD 128

typedef __attribute__((ext_vector_type(2))) float v2f;
typedef __attribute__((ext_vector_type(8))) float v8f;
typedef __attribute__((ext_vector_type(4))) int   v4i;

#if __has_builtin(__builtin_amdgcn_global_load_async_to_lds_b128)
#define USE_ASYNC_LDS 1
#else
#define USE_ASYNC_LDS 0
#endif

// 16-byte global -> LDS stage. Async (GLOBAL_LOAD_ASYNC_TO_LDS_B128, ASYNCcnt)
// when the builtin exists; otherwise a plain float4 round trip through VGPRs.
__device__ __forceinline__ void stage16(const float* __restrict__ g, float* l) {
#if USE_ASYNC_LDS
  __builtin_amdgcn_global_load_async_to_lds_b128(
      (__attribute__((address_space(1))) v4i*)g,
      (__attribute__((address_space(3))) v4i*)l, 0, 0);
#else
  *(float4*)l = *(const float4*)g;
#endif
}

__device__ __forceinline__ void stage_drain() {
#if USE_ASYNC_LDS
  asm volatile("s_wait_asynccnt 0" ::: "memory");
#endif
}

// ---- ordered-uint encoding for float atomicMax (handles negatives) ----
__device__ __forceinline__ unsigned ordf(float f) {
  unsigned u = __float_as_uint(f);
  return (u & 0x80000000u) ? ~u : (u | 0x80000000u);
}
__device__ __forceinline__ float unordf(unsigned e) {
  unsigned u = (e & 0x80000000u) ? (e ^ 0x80000000u) : ~e;
  return __uint_as_float(u);
}

// ---- zero the workspace accumulators (harness poisons ws; re-init per call) ----
__global__ void init_kernel(float* __restrict__ neigh, unsigned* __restrict__ maxbits,
                            float* __restrict__ den, int N) {
  int tid = blockIdx.x * blockDim.x + threadIdx.x;
  if (tid < N * DD) neigh[tid] = 0.0f;
  if (tid < N) { maxbits[tid] = 0u; den[tid] = 0.0f; }  // 0u encodes below all reals
}

// ---- pass 1: per-edge score = <h[src]*rel[rid], h[dst]>; segment max via atomicMax ----
// one wave32 per edge; lane l owns floats [4l, 4l+4)
__global__ void score_max_kernel(const float* __restrict__ ent, const float* __restrict__ rel,
                                 const int* __restrict__ src, const int* __restrict__ dst,
                                 const int* __restrict__ rid, float* __restrict__ score,
                                 unsigned* __restrict__ maxbits, int E) {
  int edge = blockIdx.x * 8 + (threadIdx.x >> 5);
  int lane = threadIdx.x & 31;
  if (edge >= E) return;
  int s = src[edge], d = dst[edge], r = rid[edge];
  float4 hs = ((const float4*)(ent + (size_t)s * DD))[lane];
  float4 rl = ((const float4*)(rel + (size_t)r * DD))[lane];
  float4 hd = ((const float4*)(ent + (size_t)d * DD))[lane];
  float p = hs.x * rl.x * hd.x + hs.y * rl.y * hd.y +
            hs.z * rl.z * hd.z + hs.w * rl.w * hd.w;
  // wave32 butterfly reduction
  for (int off = 16; off >= 1; off >>= 1) p += __shfl_xor(p, off, 32);
  if (lane == 0) {
    score[edge] = p;
    atomicMax(&maxbits[d], ordf(p));
  }
}

// ---- pass 2: ex = exp(score - max[dst]) (in place); den[dst] += ex ----
__global__ void exp_den_kernel(float* __restrict__ score_ex, const unsigned* __restrict__ maxbits,
                               const int* __restrict__ dst, float* __restrict__ den, int E) {
  int e = blockIdx.x * blockDim.x + threadIdx.x;
  if (e >= E) return;
  int d = dst[e];
  float ex = expf(score_ex[e] - unordf(maxbits[d]));
  score_ex[e] = ex;
  unsafeAtomicAdd(&den[d], ex);  // global_atomic_add_f32
}

// ---- pass 3: neigh[dst] += (h[src]*rel) * alpha, alpha = ex/den[dst] ----
__global__ void scatter_kernel(const float* __restrict__ ent, const float* __restrict__ rel,
                               const int* __restrict__ src, const int* __restrict__ dst,
                               const int* __restrict__ rid, const float* __restrict__ ex,
                               const float* __restrict__ den, float* __restrict__ neigh, int E) {
  int edge = blockIdx.x * 8 + (threadIdx.x >> 5);
  int lane = threadIdx.x & 31;
  if (edge >= E) return;
  int s = src[edge], d = dst[edge], r = rid[edge];
  float alpha = ex[edge] / den[d];
  float4 hs = ((const float4*)(ent + (size_t)s * DD))[lane];
  float4 rl = ((const float4*)(rel + (size_t)r * DD))[lane];
  float* base = neigh + (size_t)d * DD + lane * 4;
  unsafeAtomicAdd(base + 0, hs.x * rl.x * alpha);
  unsafeAtomicAdd(base + 1, hs.y * rl.y * alpha);
  unsafeAtomicAdd(base + 2, hs.z * rl.z * alpha);
  unsafeAtomicAdd(base + 3, hs.w * rl.w * alpha);
}

// ---- pass 4: out = tanh(neigh @ W) via V_WMMA_F32_16X16X4_F32 ----
// block = 256 threads = 8 waves; block owns 64 output rows (4 row-tiles of 16).
// W (128x128 = 64 KB) and the A slab (64x128 = 32 KB) are staged in LDS
// (96 KB of the 320 KB WGP LDS); inner loop is pure ds + wmma.
// wave w owns output cols [16w, 16w+16).
__global__ void gemm_tanh_kernel(const float* __restrict__ neigh, const float* __restrict__ W,
                                 float* __restrict__ out, int N) {
  __shared__ float Ws[DD * DD];   // 64 KB
  __shared__ float As[64 * DD];   // 32 KB
  int rowBase = blockIdx.x * 64;

  // stage W: 16384 floats, 256 threads -> 16 x b128 per thread, coalesced
  for (int i = threadIdx.x * 4; i < DD * DD; i += 256 * 4)
    stage16(&W[i], &Ws[i]);
  // stage A slab: 8192 floats -> 8 x b128 per thread
  for (int i = threadIdx.x * 4; i < 64 * DD; i += 256 * 4) {
    int row = rowBase + (i >> 7);
    if (row < N) {
      stage16(&neigh[(size_t)row * DD + (i & 127)], &As[i]);
    } else {
      As[i] = 0.0f; As[i + 1] = 0.0f; As[i + 2] = 0.0f; As[i + 3] = 0.0f;
    }
  }
  stage_drain();     // s_wait_asynccnt 0 on the async path
  __syncthreads();

  int wave = threadIdx.x >> 5;
  int lane = threadIdx.x & 31;
  int tileN = wave * 16;
  int rc = lane & 15;          // A row-in-tile / B-and-D column for this lane
  int kb = (lane >> 4) * 2;    // K sub-offset within each 4-wide K step

  for (int t = 0; t < 4; ++t) {
    const float* Arow = &As[(t * 16 + rc) * DD];
    v8f c = {};
    for (int k0 = 0; k0 < DD; k0 += 4) {
      v2f a, b;
      a.x = Arow[k0 + kb];
      a.y = Arow[k0 + kb + 1];
      b.x = Ws[(k0 + kb) * DD + tileN + rc];
      b.y = Ws[(k0 + kb + 1) * DD + tileN + rc];
      // 8 args: (neg_a, A, neg_b, B, c_mod, C, reuse_a, reuse_b)
      c = __builtin_amdgcn_wmma_f32_16x16x4_f32(false, a, false, b, (short)0, c, false, false);
    }
    // C/D layout: VGPR r -> (M = r + (lane>=16 ? 8 : 0), N = lane&15)
    int mBase = rowBase + t * 16 + ((lane >> 4) * 8);
    for (int r = 0; r < 8; ++r) {
      int m = mBase + r;
      if (m < N) out[(size_t)m * DD + tileN + rc] = tanhf(c[r]);
    }
  }
}

extern "C" void kernel_launch(void* const* d_in, const int* in_sizes, int n_in,
                              void* d_out, int out_size, void* d_ws, size_t ws_size,
                              hipStream_t stream) {
  const float* ent = (const float*)d_in[0];  // [N, D]
  const float* rel = (const float*)d_in[1];  // [R2, D]
  const float* W   = (const float*)d_in[2];  // [D, D]
  // d_in[3] = node_id (identity permutation, unused)
  const int* src = (const int*)d_in[4];      // [E]
  const int* dst = (const int*)d_in[5];      // [E]
  const int* rid = (const int*)d_in[6];      // [E]

  int N = in_sizes[0] / DD;
  int E = in_sizes[4];

  // workspace layout (floats): neigh[N*D] | score/ex[E] | maxbits[N] | den[N]
  float*    neigh   = (float*)d_ws;
  float*    score   = neigh + (size_t)N * DD;
  unsigned* maxbits = (unsigned*)(score + E);
  float*    den     = (float*)(maxbits + N);
  float*    out     = (float*)d_out;

  int initBlocks  = (N * DD + 255) / 256;
  int edgeBlocks8 = (E + 7) / 8;       // wave-per-edge kernels (8 edges / 256-thr block)
  int edgeBlocks  = (E + 255) / 256;   // thread-per-edge kernel
  int gemmBlocks  = (N + 63) / 64;     // 64 output rows per block

  init_kernel     <<<initBlocks,  256, 0, stream>>>(neigh, maxbits, den, N);
  score_max_kernel<<<edgeBlocks8, 256, 0, stream>>>(ent, rel, src, dst, rid, score, maxbits, E);
  exp_den_kernel  <<<edgeBlocks,  256, 0, stream>>>(score, maxbits, dst, den, E);
  scatter_kernel  <<<edgeBlocks8, 256, 0, stream>>>(ent, rel, src, dst, rid, score, den, neigh, E);
  gemm_tanh_kernel<<<gemmBlocks,  256, 0, stream>>>(neigh, W, out, N);
}